// BWCholeskyBlock_87153476371131
// MI455X (gfx1250) — compile-verified
//
#include <hip/hip_runtime.h>
#include <hip/hip_bf16.h>

typedef __attribute__((ext_vector_type(2))) float v2f;
typedef __attribute__((ext_vector_type(8))) float v8f;

#define BN 64
#define CC 256
#define HH 56
#define WW 56
#define HWSZ (HH * WW)          // 3136
#define GG 32
#define CSZ 8
#define MTOT (BN * HWSZ)        // 200704
#define EPSV 1e-5f
#define MOM 0.1f
#define FIXF 0.9f

// workspace layout (floats):
//  [0,256)      channel sums (atomic accum)
//  [256,2304)   gram accum [G][8][8]
//  [2304,2560)  mean [G][8]
//  [2560,4608)  winv = L^{-1} [G][8][8]
#define WS_SUMS 0
#define WS_GRAM 256
#define WS_MEAN 2304
#define WS_WINV 2560

__global__ __launch_bounds__(256) void bw_zero_kernel(float* ws) {
    int i = threadIdx.x;
    #pragma unroll
    for (int k = 0; k < 9; ++k) {
        int idx = i + k * 256;
        if (idx < 2304) ws[idx] = 0.0f;
    }
}

// Pass 1: per-group sums and Gram matrices via V_WMMA_F32_16X16X4_F32.
// grid.x = G/2 group-pairs (16 channels -> one 16x16 WMMA tile),
// grid.y = B batches. Each wave streams K=4 chunks of the m-axis; the
// f32 A(16x4) and B(4x16) WMMA layouts coincide per-lane, so one float2
// load feeds both operands of x * x^T.
__global__ __launch_bounds__(256) void bw_stats_kernel(const float* __restrict__ X,
                                                       float* __restrict__ ws) {
    const int gp   = blockIdx.x;          // 0..15
    const int b    = blockIdx.y;          // 0..63
    const int tid  = threadIdx.x;
    const int wv   = tid >> 5;            // 8 waves
    const int lane = tid & 31;
    const int r    = lane & 15;           // row/col index in 16x16 tile
    const int koff = (lane >> 4) << 1;    // K sub-offset {0,2}

    const float* rowp = X + ((size_t)b * CC + gp * 16 + r) * HWSZ;

    v8f c0 = {};
    v8f c1 = {};
    float sacc = 0.0f;

    // 3136 elements = 784 K-chunks of 4; wave handles chunks {wv, wv+8, ...}
    for (int t = wv; t < 784; t += 16) {
        const int s0 = (t << 2) + koff;
        v2f a0 = *reinterpret_cast<const v2f*>(rowp + s0);
        v2f a1 = *reinterpret_cast<const v2f*>(rowp + s0 + 32);
        sacc += a0.x + a0.y + a1.x + a1.y;
        c0 = __builtin_amdgcn_wmma_f32_16x16x4_f32(false, a0, false, a0,
                                                   (short)0, c0, false, false);
        c1 = __builtin_amdgcn_wmma_f32_16x16x4_f32(false, a1, false, a1,
                                                   (short)0, c1, false, false);
    }
    v8f cacc = c0 + c1;

    // workgroup-level reduction in LDS, then one global atomic per entry
    __shared__ float lds_gram[128];   // [2 groups][8][8]
    __shared__ float lds_sum[16];
    for (int i = tid; i < 128; i += 256) lds_gram[i] = 0.0f;
    if (tid < 16) lds_sum[tid] = 0.0f;
    __syncthreads();

    atomicAdd(&lds_sum[r], sacc);

    // C-matrix layout: VGPR j, lane L -> row M = (L<16 ? j : j+8), col N = L&15.
    // Useful lanes: 0-7 (group0 Gram cols), 24-31 (group1 Gram cols).
    if (lane < 8) {
        #pragma unroll
        for (int j = 0; j < 8; ++j)
            atomicAdd(&lds_gram[j * 8 + lane], cacc[j]);
    } else if (lane >= 24) {
        const int n = lane - 24;
        #pragma unroll
        for (int j = 0; j < 8; ++j)
            atomicAdd(&lds_gram[64 + j * 8 + n], cacc[j]);
    }
    __syncthreads();

    if (tid < 128) atomicAdd(&ws[WS_GRAM + gp * 128 + tid], lds_gram[tid]);
    if (tid < 16)  atomicAdd(&ws[WS_SUMS + gp * 16 + tid], lds_sum[tid]);
}

// Pass 2: one thread per group: finalize cov, running stats, cov-fix,
// 8x8 Cholesky, invert L; emit rm/rc outputs and whitening matrix.
__global__ __launch_bounds__(32) void bw_finalize_kernel(const float* __restrict__ rmean_in,
                                                         const float* __restrict__ rcov_in,
                                                         float* __restrict__ ws,
                                                         float* __restrict__ out_rm,
                                                         float* __restrict__ out_rc) {
    const int g = threadIdx.x;
    if (g >= GG) return;

    const float invM = 1.0f / (float)MTOT;
    float mean[8];
    float A[8][8];

    #pragma unroll
    for (int j = 0; j < 8; ++j) mean[j] = ws[WS_SUMS + g * 8 + j] * invM;

    #pragma unroll
    for (int j = 0; j < 8; ++j) {
        #pragma unroll
        for (int k = 0; k < 8; ++k) {
            float cv = ws[WS_GRAM + g * 64 + j * 8 + k] * invM - mean[j] * mean[k];
            if (j == k) cv += EPSV;
            float rc = (1.0f - MOM) * rcov_in[g * 64 + j * 8 + k] + MOM * cv;
            float rcf = (j == k) ? rc : FIXF * rc;
            A[j][k] = rcf;
            out_rc[g * 64 + j * 8 + k] = rcf;
        }
    }
    #pragma unroll
    for (int j = 0; j < 8; ++j) {
        out_rm[g * 8 + j] = (1.0f - MOM) * rmean_in[g * 8 + j] + MOM * mean[j];
        ws[WS_MEAN + g * 8 + j] = mean[j];
    }

    // Cholesky (lower), in place
    #pragma unroll
    for (int k = 0; k < 8; ++k) {
        float d = A[k][k];
        #pragma unroll
        for (int p = 0; p < 8; ++p) if (p < k) d -= A[k][p] * A[k][p];
        d = sqrtf(d);
        A[k][k] = d;
        float rd = 1.0f / d;
        #pragma unroll
        for (int i = 0; i < 8; ++i) {
            if (i > k) {
                float s = A[i][k];
                #pragma unroll
                for (int p = 0; p < 8; ++p) if (p < k) s -= A[i][p] * A[k][p];
                A[i][k] = s * rd;
            }
        }
    }

    // Winv = L^{-1} (lower triangular)
    float Wi[8][8];
    #pragma unroll
    for (int j = 0; j < 8; ++j) {
        #pragma unroll
        for (int i = 0; i < 8; ++i) Wi[i][j] = 0.0f;
        Wi[j][j] = 1.0f / A[j][j];
        #pragma unroll
        for (int i = 0; i < 8; ++i) {
            if (i > j) {
                float s = 0.0f;
                #pragma unroll
                for (int p = 0; p < 8; ++p)
                    if (p >= j && p < i) s += A[i][p] * Wi[p][j];
                Wi[i][j] = -s / A[i][i];
            }
        }
    }
    #pragma unroll
    for (int i = 0; i < 8; ++i)
        #pragma unroll
        for (int j = 0; j < 8; ++j)
            ws[WS_WINV + g * 64 + i * 8 + j] = Wi[i][j];
}

// Pass 3: Y = Winv * (x - mean) + beta. grid = (m/256, G). Coalesced along W.
__global__ __launch_bounds__(256) void bw_apply_kernel(const float* __restrict__ X,
                                                       const float* __restrict__ ws,
                                                       const float* __restrict__ beta,
                                                       float* __restrict__ Y) {
    const int g = blockIdx.y;
    const int tid = threadIdx.x;

    __shared__ float sm[8];
    __shared__ float sb[8];
    __shared__ float sw[64];
    if (tid < 8)  { sm[tid] = ws[WS_MEAN + g * 8 + tid]; sb[tid] = beta[g * 8 + tid]; }
    if (tid < 64) sw[tid] = ws[WS_WINV + g * 64 + tid];
    __syncthreads();

    const int m = blockIdx.x * 256 + tid;        // 784*256 == MTOT exactly
    const int b = m / HWSZ;
    const int s = m - b * HWSZ;
    const size_t base = ((size_t)b * CC + g * 8) * HWSZ + s;

    float xc[8];
    #pragma unroll
    for (int j = 0; j < 8; ++j) xc[j] = X[base + (size_t)j * HWSZ] - sm[j];

    #pragma unroll
    for (int i = 0; i < 8; ++i) {
        float y = sb[i];
        #pragma unroll
        for (int j = 0; j < 8; ++j)
            if (j <= i) y = fmaf(sw[i * 8 + j], xc[j], y);
        Y[base + (size_t)i * HWSZ] = y;
    }
}

extern "C" void kernel_launch(void* const* d_in, const int* in_sizes, int n_in,
                              void* d_out, int out_size, void* d_ws, size_t ws_size,
                              hipStream_t stream) {
    const float* X     = (const float*)d_in[0];
    const float* rmean = (const float*)d_in[1];
    const float* rcov  = (const float*)d_in[2];
    const float* beta  = (const float*)d_in[3];

    float* out   = (float*)d_out;
    float* outY  = out;
    float* outRM = out + in_sizes[0];                  // Y count = B*C*H*W
    float* outRC = out + in_sizes[0] + in_sizes[1];    // after running_mean
    float* ws    = (float*)d_ws;

    bw_zero_kernel<<<1, 256, 0, stream>>>(ws);

    dim3 g1(16, 64);
    bw_stats_kernel<<<g1, 256, 0, stream>>>(X, ws);

    bw_finalize_kernel<<<1, 32, 0, stream>>>(rmean, rcov, ws, outRM, outRC);

    dim3 g3(MTOT / 256, GG);
    bw_apply_kernel<<<g3, 256, 0, stream>>>(X, ws, beta, outY);
}